// MultiGCN_38860864094260
// MI455X (gfx1250) — compile-verified
//
#include <hip/hip_runtime.h>
#include <math.h>

// ---------------------------------------------------------------------------
// MultiGCN forward for MI455X (gfx1250, wave32, WMMA).
// All heavy math runs through V_WMMA_F32_16X16X4_F32 (fp32 tensor path):
// precision is required because the kNN selection thresholds exp(-d2/9).
// Workspace layout (floats): R0[N*N] | R1[N*N] | R2[N*D] | sq[N] thr[N]
// invd[N] mean[D] istd[D]  => ~168 MB.
// ---------------------------------------------------------------------------

typedef __attribute__((ext_vector_type(2))) float v2f;
typedef __attribute__((ext_vector_type(8))) float v8f;

#define GN 4096      // number of nodes
#define GD 2048      // input feature dim
#define GH 1000      // hidden dim
#define BN_EPS 1e-5f

enum { EPI_PLAIN = 0, EPI_ADJ = 1, EPI_BIAS = 2 };

// ---------------------------------------------------------------------------
// C[M,Nn] = A[M,K] @ (TRANSB ? B[Nn,K]^T : B[K,Nn])   (all row-major fp32)
// Block tile 128(M) x 64(N), K-panel 16. 8 waves; each wave owns 32x32 via
// four 16x16 f32 WMMA accumulators. Fragment layouts follow the CDNA5 ISA:
//   A 16x4 : a[j] = A[lane%16][2*(lane/16)+j]
//   B 4x16 : b[j] = B[2*(lane/16)+j][lane%16]
//   C 16x16: c[r] -> row = r + 8*(lane/16), col = lane%16
// Epilogues: EPI_ADJ  -> exp(-max(sq[r]+sq[c]-2*acc, 0)/9)   (aux = row norms)
//            EPI_BIAS -> acc + aux[col]
// ---------------------------------------------------------------------------
template<bool TRANSB, int EPI>
__global__ __launch_bounds__(256)
void wmma_gemm_kernel(const float* __restrict__ A, const float* __restrict__ B,
                      float* __restrict__ C, int M, int Nn, int K,
                      const float* __restrict__ aux)
{
    __shared__ float As[128][20];   // 128x16, padded to kill bank conflicts
    __shared__ float Bs[16][68];    // 16x64, padded

    const int tid  = threadIdx.x;
    const int lane = tid & 31;
    const int wid  = tid >> 5;      // 0..7
    const int wm   = wid & 3;       // wave M index (4 x 32 rows)
    const int wn   = wid >> 2;      // wave N index (2 x 32 cols)
    const int lr   = lane & 15;
    const int khi  = lane >> 4;     // 0/1
    const int m0   = blockIdx.y * 128;
    const int n0   = blockIdx.x * 64;

    v8f acc[2][2];
    #pragma unroll
    for (int i = 0; i < 2; ++i)
        #pragma unroll
        for (int j = 0; j < 2; ++j)
            #pragma unroll
            for (int e = 0; e < 8; ++e) acc[i][j][e] = 0.0f;

    for (int k0 = 0; k0 < K; k0 += 16) {
        // ---- stage A tile: 128x16, 256 thr x 2 float4 each ----
        {
            const int q = tid & 3;
            const int r = tid >> 2;           // 0..63
            #pragma unroll
            for (int p = 0; p < 2; ++p) {
                const int row = r + p * 64;
                const float4 v = *reinterpret_cast<const float4*>(
                    A + (size_t)(m0 + row) * K + k0 + q * 4);
                *reinterpret_cast<float4*>(&As[row][q * 4]) = v;
            }
        }
        // ---- stage B tile: 16x64 ----
        if (!TRANSB) {
            const int q   = tid & 15;         // column group
            const int r   = tid >> 4;         // k row 0..15
            const int col = n0 + q * 4;
            const float* src = B + (size_t)(k0 + r) * Nn;
            if (col + 3 < Nn) {
                const float4 v = *reinterpret_cast<const float4*>(src + col);
                *reinterpret_cast<float4*>(&Bs[r][q * 4]) = v;
            } else {
                #pragma unroll
                for (int j = 0; j < 4; ++j) {
                    int c = col + j; if (c > Nn - 1) c = Nn - 1;  // clamp; store masked
                    Bs[r][q * 4 + j] = src[c];
                }
            }
        } else {
            // gram path only (Nn multiple of 64): read B row-major, transpose to LDS
            const int q = tid & 3;
            const int r = tid >> 2;           // n 0..63
            const float4 v = *reinterpret_cast<const float4*>(
                B + (size_t)(n0 + r) * K + k0 + q * 4);
            Bs[q * 4 + 0][r] = v.x;
            Bs[q * 4 + 1][r] = v.y;
            Bs[q * 4 + 2][r] = v.z;
            Bs[q * 4 + 3][r] = v.w;
        }
        __syncthreads();

        // ---- 4 WMMA K-steps of 4 ----
        #pragma unroll
        for (int kk = 0; kk < 4; ++kk) {
            v2f af[2], bf[2];
            #pragma unroll
            for (int i = 0; i < 2; ++i) {
                const int mr = wm * 32 + i * 16 + lr;
                af[i][0] = As[mr][kk * 4 + 2 * khi + 0];
                af[i][1] = As[mr][kk * 4 + 2 * khi + 1];
            }
            #pragma unroll
            for (int j = 0; j < 2; ++j) {
                const int nc = wn * 32 + j * 16 + lr;
                bf[j][0] = Bs[kk * 4 + 2 * khi + 0][nc];
                bf[j][1] = Bs[kk * 4 + 2 * khi + 1][nc];
            }
            #pragma unroll
            for (int i = 0; i < 2; ++i)
                #pragma unroll
                for (int j = 0; j < 2; ++j)
                    acc[i][j] = __builtin_amdgcn_wmma_f32_16x16x4_f32(
                        false, af[i], false, bf[j], (short)0, acc[i][j],
                        false, false);
        }
        __syncthreads();
    }

    // ---- epilogue + store (predicated on valid columns) ----
    #pragma unroll
    for (int i = 0; i < 2; ++i) {
        #pragma unroll
        for (int j = 0; j < 2; ++j) {
            const int col = n0 + wn * 32 + j * 16 + lr;
            if (col < Nn) {
                #pragma unroll
                for (int r2 = 0; r2 < 8; ++r2) {
                    const int row = m0 + wm * 32 + i * 16 + khi * 8 + r2;
                    float v = acc[i][j][r2];
                    if (EPI == EPI_ADJ) {
                        float d2 = aux[row] + aux[col] - 2.0f * v;
                        d2 = d2 > 0.0f ? d2 : 0.0f;
                        v = __expf(d2 * (-1.0f / 9.0f));
                    } else if (EPI == EPI_BIAS) {
                        v += aux[col];
                    }
                    C[(size_t)row * Nn + col] = v;
                }
            }
        }
    }
}

// ---------------------------------------------------------------------------
// sq[row] = sum_c x[row][c]^2
// ---------------------------------------------------------------------------
__global__ __launch_bounds__(256)
void rownorm_kernel(const float* __restrict__ x, int cols, float* __restrict__ sq)
{
    __shared__ float red[256];
    const int row = blockIdx.x;
    const float* r = x + (size_t)row * cols;
    float s = 0.0f;
    for (int c = threadIdx.x; c < cols; c += 256) { const float v = r[c]; s += v * v; }
    red[threadIdx.x] = s; __syncthreads();
    for (int off = 128; off > 0; off >>= 1) {
        if (threadIdx.x < off) red[threadIdx.x] += red[threadIdx.x + off];
        __syncthreads();
    }
    if (threadIdx.x == 0) sq[row] = red[0];
}

// ---------------------------------------------------------------------------
// Per-row k-th-largest threshold via binary search over float bit patterns.
// adj_e is in (0,1] (positive floats -> uint order == float order).
// Invariant: count(x >= lo) >= k, count(x >= hi) < k.  One block per row.
// ---------------------------------------------------------------------------
__global__ __launch_bounds__(256)
void topk_threshold_kernel(const float* __restrict__ adj,
                           const int* __restrict__ nway,
                           float* __restrict__ thr)
{
    __shared__ unsigned sv[GN];
    __shared__ int red[256];
    const int row = blockIdx.x;
    const float* r = adj + (size_t)row * GN;
    for (int j = threadIdx.x; j < GN; j += 256) sv[j] = __float_as_uint(r[j]);
    __syncthreads();
    const int k = __float2int_rn((float)GN / (float)nway[0]);  // round(N/N_way)
    unsigned lo = 0u, hi = 0xFFFFFFFFu;
    for (int it = 0; it < 32; ++it) {
        const unsigned mid = lo + ((hi - lo) >> 1);
        int c = 0;
        for (int j = threadIdx.x; j < GN; j += 256) c += (sv[j] >= mid) ? 1 : 0;
        red[threadIdx.x] = c; __syncthreads();
        for (int off = 128; off > 0; off >>= 1) {
            if (threadIdx.x < off) red[threadIdx.x] += red[threadIdx.x + off];
            __syncthreads();
        }
        const int total = red[0];
        __syncthreads();
        if (total >= k) lo = mid; else hi = mid;
        if (lo + 1 >= hi) break;   // uniform across block
    }
    if (threadIdx.x == 0) thr[row] = __uint_as_float(lo);
}

// ---------------------------------------------------------------------------
// In place: adjacency = eye + mutual-kNN-mask * adj_e ; invd = 1/sqrt(rowsum+1)
// adj_e is symmetric, so mutual mask == (e >= max(thr_i, thr_j)).
// ---------------------------------------------------------------------------
__global__ __launch_bounds__(256)
void mask_rowsum_kernel(float* __restrict__ adj, const float* __restrict__ thr,
                        float* __restrict__ invd)
{
    __shared__ float red[256];
    const int row = blockIdx.x;
    float* r = adj + (size_t)row * GN;
    const float ti = thr[row];
    float s = 0.0f;
    for (int j = threadIdx.x; j < GN; j += 256) {
        float v;
        if (j == row) v = 1.0f;
        else {
            const float e = r[j];
            v = (e >= ti && e >= thr[j]) ? e : 0.0f;
        }
        r[j] = v;
        s += v;
    }
    red[threadIdx.x] = s; __syncthreads();
    for (int off = 128; off > 0; off >>= 1) {
        if (threadIdx.x < off) red[threadIdx.x] += red[threadIdx.x + off];
        __syncthreads();
    }
    if (threadIdx.x == 0) invd[row] = rsqrtf(red[0] + 1.0f);
}

// adj[i][j] *= invd[i]*invd[j]
__global__ __launch_bounds__(256)
void sym_norm_kernel(float* __restrict__ adj, const float* __restrict__ invd)
{
    const size_t total = (size_t)GN * GN;
    const size_t step  = (size_t)gridDim.x * blockDim.x;
    for (size_t i = (size_t)blockIdx.x * blockDim.x + threadIdx.x; i < total; i += step) {
        const int row = (int)(i >> 12);
        const int col = (int)(i & (GN - 1));
        adj[i] *= invd[row] * invd[col];
    }
}

// a_out (holds An@An) <- w0*I + w1*An + w2*(An@An); softmax of aifa on device
__global__ __launch_bounds__(256)
void blend_kernel(float* __restrict__ a_out, const float* __restrict__ an,
                  const float* __restrict__ p1, const float* __restrict__ p2,
                  const float* __restrict__ p3)
{
    const float x1 = p1[0], x2 = p2[0], x3 = p3[0];
    const float mx = fmaxf(x1, fmaxf(x2, x3));
    const float e1 = __expf(x1 - mx), e2 = __expf(x2 - mx), e3 = __expf(x3 - mx);
    const float inv = 1.0f / (e1 + e2 + e3);
    const float w0 = e1 * inv, w1 = e2 * inv, w2 = e3 * inv;
    const size_t total = (size_t)GN * GN;
    const size_t step  = (size_t)gridDim.x * blockDim.x;
    for (size_t i = (size_t)blockIdx.x * blockDim.x + threadIdx.x; i < total; i += step) {
        const int row = (int)(i >> 12);
        const int col = (int)(i & (GN - 1));
        a_out[i] = w1 * an[i] + w2 * a_out[i] + (row == col ? w0 : 0.0f);
    }
}

// Training-mode BN stats over rows (biased variance); coalesced over columns.
__global__ __launch_bounds__(256)
void bn_stats_kernel(const float* __restrict__ x, int rows, int cols,
                     float* __restrict__ mean, float* __restrict__ istd)
{
    const int c = blockIdx.x * blockDim.x + threadIdx.x;
    if (c >= cols) return;
    float s = 0.0f, s2 = 0.0f;
    for (int r = 0; r < rows; ++r) {
        const float v = x[(size_t)r * cols + c];
        s += v; s2 += v * v;
    }
    const float m = s / (float)rows;
    float var = s2 / (float)rows - m * m;
    var = var > 0.0f ? var : 0.0f;
    mean[c] = m;
    istd[c] = rsqrtf(var + BN_EPS);
}

__global__ __launch_bounds__(256)
void bn_apply_relu_kernel(const float* __restrict__ x, int cols, size_t total,
                          const float* __restrict__ mean, const float* __restrict__ istd,
                          const float* __restrict__ gamma, const float* __restrict__ beta,
                          float* __restrict__ out)
{
    const size_t step = (size_t)gridDim.x * blockDim.x;
    for (size_t i = (size_t)blockIdx.x * blockDim.x + threadIdx.x; i < total; i += step) {
        const int c = (int)(i % (size_t)cols);
        const float v = (x[i] - mean[c]) * istd[c] * gamma[c] + beta[c];
        out[i] = fmaxf(v, 0.0f);
    }
}

// ---------------------------------------------------------------------------
extern "C" void kernel_launch(void* const* d_in, const int* in_sizes, int n_in,
                              void* d_out, int out_size, void* d_ws, size_t ws_size,
                              hipStream_t stream)
{
    (void)in_sizes; (void)n_in; (void)out_size; (void)ws_size;

    const float* features  = (const float*)d_in[0];
    const float* bn1_gamma = (const float*)d_in[1];
    const float* bn1_beta  = (const float*)d_in[2];
    const float* bn2_gamma = (const float*)d_in[3];
    const float* bn2_beta  = (const float*)d_in[4];
    const float* gcn_w     = (const float*)d_in[5];
    const float* gcn_b     = (const float*)d_in[6];
    const float* aifa1     = (const float*)d_in[7];
    const float* aifa2     = (const float*)d_in[8];
    const float* aifa3     = (const float*)d_in[9];
    const int*   nway      = (const int*)d_in[10];
    float* out = (float*)d_out;

    const size_t NNp = (size_t)GN * GN;
    float* R0   = (float*)d_ws;               // [N,N]  adj_e / An
    float* R1   = R0 + NNp;                   // [N,N]  An@An -> blended A
    float* R2   = R1 + NNp;                   // [N,D]  x (and later y[N,H])
    float* sq   = R2 + (size_t)GN * GD;       // [N]
    float* thr  = sq + GN;                    // [N]
    float* invd = thr + GN;                   // [N]
    float* mean = invd + GN;                  // [D]
    float* istd = mean + GD;                  // [D]

    const dim3 blk(256);
    const dim3 gNN((GN + 63) / 64, GN / 128);
    const dim3 gND((GD + 63) / 64, GN / 128);
    const dim3 gNH((GH + 63) / 64, GN / 128);
    const int ELEM_BLOCKS = 4096;

    // ---------------- stage 1: A1 = multi_adjacency(features) ----------------
    rownorm_kernel<<<GN, blk, 0, stream>>>(features, GD, sq);
    wmma_gemm_kernel<true, EPI_ADJ><<<gNN, blk, 0, stream>>>(
        features, features, R0, GN, GN, GD, sq);                     // adj_e
    topk_threshold_kernel<<<GN, blk, 0, stream>>>(R0, nway, thr);
    mask_rowsum_kernel<<<GN, blk, 0, stream>>>(R0, thr, invd);
    sym_norm_kernel<<<ELEM_BLOCKS, blk, 0, stream>>>(R0, invd);      // An
    wmma_gemm_kernel<false, EPI_PLAIN><<<gNN, blk, 0, stream>>>(
        R0, R0, R1, GN, GN, GN, nullptr);                            // An@An
    blend_kernel<<<ELEM_BLOCKS, blk, 0, stream>>>(R1, R0, aifa1, aifa2, aifa3);

    // x = relu(bn1(A1 @ features))
    wmma_gemm_kernel<false, EPI_PLAIN><<<gND, blk, 0, stream>>>(
        R1, features, R2, GN, GD, GN, nullptr);
    bn_stats_kernel<<<(GD + 255) / 256, blk, 0, stream>>>(R2, GN, GD, mean, istd);
    bn_apply_relu_kernel<<<ELEM_BLOCKS, blk, 0, stream>>>(
        R2, GD, (size_t)GN * GD, mean, istd, bn1_gamma, bn1_beta, R2);

    // ---------------- stage 2: A2 = multi_adjacency(x) ----------------
    rownorm_kernel<<<GN, blk, 0, stream>>>(R2, GD, sq);
    wmma_gemm_kernel<true, EPI_ADJ><<<gNN, blk, 0, stream>>>(
        R2, R2, R0, GN, GN, GD, sq);
    topk_threshold_kernel<<<GN, blk, 0, stream>>>(R0, nway, thr);
    mask_rowsum_kernel<<<GN, blk, 0, stream>>>(R0, thr, invd);
    sym_norm_kernel<<<ELEM_BLOCKS, blk, 0, stream>>>(R0, invd);
    wmma_gemm_kernel<false, EPI_PLAIN><<<gNN, blk, 0, stream>>>(
        R0, R0, R1, GN, GN, GN, nullptr);
    blend_kernel<<<ELEM_BLOCKS, blk, 0, stream>>>(R1, R0, aifa1, aifa2, aifa3);

    // ---------------- stage 3: out = relu(bn2(A2 @ (x @ W) + b)) ----------------
    wmma_gemm_kernel<false, EPI_PLAIN><<<gNH, blk, 0, stream>>>(
        R2, gcn_w, out, GN, GH, GD, nullptr);                        // xw -> d_out (scratch)
    wmma_gemm_kernel<false, EPI_BIAS><<<gNH, blk, 0, stream>>>(
        R1, out, R2, GN, GH, GN, gcn_b);                             // y -> R2
    bn_stats_kernel<<<(GH + 255) / 256, blk, 0, stream>>>(R2, GN, GH, mean, istd);
    bn_apply_relu_kernel<<<ELEM_BLOCKS, blk, 0, stream>>>(
        R2, GH, (size_t)GN * GH, mean, istd, bn2_gamma, bn2_beta, out);
}